// GaussianBasedGraphAggregationMaskGenerator_3496103379386
// MI455X (gfx1250) — compile-verified
//
#include <hip/hip_runtime.h>
#include <math.h>

typedef __attribute__((ext_vector_type(2))) float v2f;
typedef __attribute__((ext_vector_type(8))) float v8f;

#define NMAX 64
#define GP_JITTER 1e-5f

__global__ __launch_bounds__(32)
void gp_gain_mask_kernel(const float* __restrict__ t, const int* __restrict__ vid,
                         const float* __restrict__ noise, const float* __restrict__ outscale,
                         const float* __restrict__ lscale, const float* __restrict__ alpha,
                         float* __restrict__ outMask, float* __restrict__ outG,
                         int B, int L, int V, int Kn)
{
    __shared__ float Cl[NMAX * NMAX];   // Kn (noisy) -> Cholesky factor (lower), in place
    __shared__ float Ml[NMAX * NMAX];   // Kxx -> V = C^{-1} Kxx, in place
    __shared__ float tU[NMAX];
    __shared__ int   iU[NMAX];
    __shared__ float sT[NMAX];
    __shared__ int   sI[NMAX];
    __shared__ float sums[NMAX];
    __shared__ float gn[NMAX];
    __shared__ float cg[NMAX];
    __shared__ int   grp[NMAX];
    __shared__ int   cnt;

    const int lane = threadIdx.x;
    const int b = blockIdx.x / V;
    const int v = blockIdx.x % V;
    const float nz = noise[v];
    const float os = outscale[v];
    const float ls = lscale[v];
    const float al = alpha[v];

    // ---- Phase 0: gather this variable's points, then stable rank-sort by (time, idx)
    if (lane == 0) cnt = 0;
    __syncthreads();
    for (int i = lane; i < L; i += 32) {
        if (vid[b * L + i] == v) {
            int p = atomicAdd(&cnt, 1);
            if (p < NMAX) { tU[p] = t[b * L + i]; iU[p] = i; }
        }
    }
    __syncthreads();
    int n = cnt; if (n > NMAX) n = NMAX;
    const int nb = (n + 15) >> 4;
    const int npad = nb << 4;

    for (int e = lane; e < n; e += 32) {
        float te = tU[e]; int ie = iU[e]; int r = 0;
        for (int q = 0; q < n; ++q) {
            float tq = tU[q]; int iq = iU[q];
            r += (tq < te) || (tq == te && iq < ie);
        }
        sT[r] = te; sI[r] = ie;
    }
    __syncthreads();

    // ---- Phase 1: build RQ kernel matrices (padded region: Ml=0, Cl=I)
    const float inv2al2 = 1.0f / (2.0f * al * ls * ls);
    for (int i = 0; i < npad; ++i) {
        for (int j = lane; j < npad; j += 32) {
            float kij = 0.0f, cij;
            if (i < n && j < n) {
                float d = sT[i] - sT[j];
                kij = os * powf(1.0f + d * d * inv2al2, -al);
                cij = (i == j) ? (os + nz + GP_JITTER) : kij;
            } else {
                cij = (i == j) ? 1.0f : 0.0f;
            }
            Ml[i * NMAX + j] = kij;
            Cl[i * NMAX + j] = cij;
        }
    }
    __syncthreads();

    // ---- Phase 2: in-place Cholesky (lower triangle) of Cl
    for (int k = 0; k < npad; ++k) {
        if (lane == 0) Cl[k * NMAX + k] = sqrtf(Cl[k * NMAX + k]);
        __syncthreads();
        float dk = Cl[k * NMAX + k];
        for (int i = k + 1 + lane; i < npad; i += 32) Cl[i * NMAX + k] /= dk;
        __syncthreads();
        for (int i = k + 1; i < npad; ++i) {
            float lik = Cl[i * NMAX + k];
            for (int j = k + 1 + lane; j <= i; j += 32)
                Cl[i * NMAX + j] -= lik * Cl[j * NMAX + k];
        }
        __syncthreads();
    }

    // ---- Phase 3: blocked forward substitution Ml <- Cl^{-1} Ml.
    // Off-diagonal block update V[ib] -= C[ib][rb] * V[rb] done with
    // V_WMMA_F32_16X16X4_F32 (4 chained K-slices per 16x16x16 product).
    const int h = lane >> 4;   // lane half selects K pair / M upper rows
    const int m = lane & 15;
    for (int ib = 0; ib < nb; ++ib) {
        const int r0 = ib << 4;
        for (int jb = 0; jb < nb; ++jb) {
            const int c0 = jb << 4;
            v8f acc;
#pragma unroll
            for (int p = 0; p < 8; ++p) acc[p] = Ml[(r0 + p + 8 * h) * NMAX + c0 + m];
            for (int rb = 0; rb < ib; ++rb) {
                const int k0c = rb << 4;
#pragma unroll
                for (int s = 0; s < 4; ++s) {
                    const int kk = 4 * s + 2 * h;
                    v2f a, bb;
                    // A = -C block: lane=M row, (vgpr,half)->K; f32 WMMA has no A-neg, so negate on load
                    a.x = -Cl[(r0 + m) * NMAX + k0c + kk];
                    a.y = -Cl[(r0 + m) * NMAX + k0c + kk + 1];
                    // B = V[rb] tile: lane=N col, (vgpr,half)->K
                    bb.x = Ml[(k0c + kk) * NMAX + c0 + m];
                    bb.y = Ml[(k0c + kk + 1) * NMAX + c0 + m];
                    acc = __builtin_amdgcn_wmma_f32_16x16x4_f32(
                        false, a, false, bb, (short)0, acc, false, false);
                }
            }
            __syncthreads();
#pragma unroll
            for (int p = 0; p < 8; ++p) Ml[(r0 + p + 8 * h) * NMAX + c0 + m] = acc[p];
            __syncthreads();
        }
        // diagonal-block triangular solve across all npad columns
        for (int r = 0; r < 16; ++r) {
            const int i = r0 + r;
            const float cii = Cl[i * NMAX + i];
            for (int j = lane; j < npad; j += 32) {
                float val = Ml[i * NMAX + j];
                for (int s2 = 0; s2 < r; ++s2)
                    val -= Cl[i * NMAX + r0 + s2] * Ml[(r0 + s2) * NMAX + j];
                Ml[i * NMAX + j] = val / cii;
            }
            __syncthreads();
        }
    }

    // ---- Phase 4: prefix variance -> posterior std sums per added observation
    {
        float c0a = 0.0f, c1a = 0.0f;
        const int j0 = lane, j1 = lane + 32;
        for (int i = 0; i < n; ++i) {
            float part = 0.0f;
            if (j0 < n) { float x = Ml[i * NMAX + j0]; c0a += x * x; part += sqrtf(fmaxf(os - c0a, 1e-12f)); }
            if (j1 < n) { float x = Ml[i * NMAX + j1]; c1a += x * x; part += sqrtf(fmaxf(os - c1a, 1e-12f)); }
            for (int off = 16; off > 0; off >>= 1) part += __shfl_xor(part, off, 32);
            if (lane == 0) sums[i] = part;
        }
    }
    __syncthreads();

    // ---- Phase 5: gains, cumulative-info K-way partition
    if (lane == 0) {
        float prev = sqrtf(os) * (float)n;
        float c = 0.0f;
        for (int i = 0; i < n; ++i) {
            float g = fmaxf(prev - sums[i], 0.0f);
            c += g;
            gn[i] = g; cg[i] = c;
            prev = sums[i];
        }
        float tot = fmaxf(c, 1e-12f);
        for (int i = 0; i < n; ++i) {
            float frac = (cg[i] - 0.5f * gn[i]) / tot;
            int g2 = (int)floorf(frac * (float)Kn);
            if (g2 < 0) g2 = 0;
            if (g2 > Kn - 1) g2 = Kn - 1;
            grp[i] = g2;
        }
    }
    __syncthreads();

    // ---- Phase 6: scatter. Each (b, point) row is owned by exactly one (b,v) block,
    // so write the full V*Kn row (zeros + one hot) and the gain.
    const int VK = V * Kn;
    for (int i = 0; i < n; ++i) {
        const int orig = sI[i];
        float* row = outMask + ((size_t)(b * L + orig)) * (size_t)VK;
        const int hot = v * Kn + grp[i];
        for (int c2 = lane; c2 < VK; c2 += 32) row[c2] = (c2 == hot) ? 1.0f : 0.0f;
        if (lane == 0) outG[b * L + orig] = gn[i];
    }
}

extern "C" void kernel_launch(void* const* d_in, const int* in_sizes, int n_in,
                              void* d_out, int out_size, void* d_ws, size_t ws_size,
                              hipStream_t stream) {
    (void)n_in; (void)d_ws; (void)ws_size;
    const float* t        = (const float*)d_in[0];
    // d_in[1] = y, unused (posterior variance does not depend on y)
    const int*   vid      = (const int*)d_in[2];
    const float* noise    = (const float*)d_in[3];
    const float* outsc    = (const float*)d_in[4];
    const float* lsc      = (const float*)d_in[5];
    const float* alp      = (const float*)d_in[6];

    const int L = 512;
    const int B = in_sizes[0] / L;
    const int V = in_sizes[3];
    // out = [B*L*V*Kn mask] ++ [B*L gains]  =>  Kn = (out_size/(B*L) - 1)/V
    const int Kn = (out_size / (B * L) - 1) / V;

    float* outMask = (float*)d_out;
    float* outG    = outMask + (size_t)B * L * V * Kn;

    dim3 grid(B * V), block(32);
    hipLaunchKernelGGL(gp_gain_mask_kernel, grid, block, 0, stream,
                       t, vid, noise, outsc, lsc, alp, outMask, outG, B, L, V, Kn);
}